// LGE_Bilinear_14139032338776
// MI455X (gfx1250) — compile-verified
//
#include <hip/hip_runtime.h>

// ---------------------------------------------------------------------------
// Complex bilinear gauge-equivariant layer for MI455X (gfx1250, wave32).
//
//   out[i,x,n,m] = sum_{j,k,r} w[i,j,k] * f1[j,x,n,r] * f2[k,x,r,m]   (complex)
//
// One real GEMM:  C[16, 9*S] = A'[16,128] @ B'[128, 9*S]
//   A' rows 0..7  = [ w_re | -w_im ],  rows 8..15 = [ w_im | w_re ]  (constant)
//   B' rows 0..63 = P_re[jk, col], rows 64..127 = P_im[jk, col],
//     P[jk, col=(x,n,m)] = sum_r f1[j,x,n,r]*f2[k,x,r,m]  (complex, stage 1)
//   C rows 0..7 = out_re[i], rows 8..15 = out_im[i].
//
// Stage 1 is computed per-lane directly in the V_WMMA_F32_16X16X4_F32
// B-fragment layout (lane = column, lane-half g owns K%4 in {2g,2g+1}), with
// v_pk_fma_f32 packed over the two B-frag members (same j, k and k+1) so the
// results bre[t]/bim[t] feed WMMA with no register shuffling.
// Stage 2 = 32 x v_wmma_f32_16x16x4_f32 per 16-column tile (K=128, fp32 exact).
// Each wave processes TPW consecutive tiles so the constant A' fragments
// (64 VGPRs, loaded from w once) are amortized. No LDS, no barriers.
// ---------------------------------------------------------------------------

typedef __attribute__((ext_vector_type(2))) float v2f;
typedef __attribute__((ext_vector_type(8))) float v8f;

#define NSITES   131072                 // 16*16*16*32 lattice sites
#define FSTRIDE  (NSITES * 9)           // per-j / per-k feature stride (floats)
#define NCOLS    (NSITES * 9)           // GEMM N = 1,179,648 columns
#define NTILES   (NCOLS / 16)           // 73,728 column tiles of 16
#define WPB      8                      // waves per block (256 threads)
#define TPW      4                      // tiles per wave

__global__ __launch_bounds__(256)
void lge_bilinear_wmma(const float* __restrict__ f1re, const float* __restrict__ f1im,
                       const float* __restrict__ f2re, const float* __restrict__ f2im,
                       const float* __restrict__ wre,  const float* __restrict__ wim,
                       float* __restrict__ out)
{
    const int lane = threadIdx.x & 31;
    const int wave = threadIdx.x >> 5;
    const int g    = lane >> 4;                 // lane half: owns K%4 in {2g,2g+1}
    const int r16  = lane & 15;                 // A-matrix row / C-matrix column
    const int tile0 = (blockIdx.x * WPB + wave) * TPW;

    // ---- A fragments: constant A'[16 x 128] from w, 32 K-steps of 4 --------
    // A 16x4 layout: lane = M; VGPR0 holds K = 2*(lane/16), VGPR1 holds K+1.
    v2f af[32];
    {
        const float* w0 = (r16 < 8) ? wre : wim;   // K <  64 half
        const float* w1 = (r16 < 8) ? wim : wre;   // K >= 64 half
        const float  s1 = (r16 < 8) ? -1.0f : 1.0f;
        const int    row = (r16 & 7) * 64;         // w[i, jk] base
#pragma unroll
        for (int t = 0; t < 16; ++t) {
            const int jk = 4 * t + 2 * g;
            af[t]      = (v2f){      w0[row + jk],      w0[row + jk + 1] };
            af[t + 16] = (v2f){ s1 * w1[row + jk], s1 * w1[row + jk + 1] };
        }
    }

    for (int tt = 0; tt < TPW; ++tt) {
        // ---- My GEMM column: (site x, output element n,m) ------------------
        const int c  = (tile0 + tt) * 16 + r16;   // global column
        const int x  = c / 9;
        const int nm = c - 9 * x;
        const int n  = nm / 3;
        const int m  = nm - 3 * n;

        // f1[j, x, n, r] for all j, r  (48 regs)
        float a_re[8][3], a_im[8][3];
#pragma unroll
        for (int j = 0; j < 8; ++j) {
            const int o = j * FSTRIDE + x * 9 + n * 3;
#pragma unroll
            for (int r = 0; r < 3; ++r) {
                a_re[j][r] = f1re[o + r];
                a_im[j][r] = f1im[o + r];
            }
        }

        // ---- Stage 1: B fragments, packed over the two frag members --------
        // Lane half g covers k in {2g, 2g+1, 4+2g, 4+2g+1}; member pair for
        // WMMA step t = 2j + kk is (k0 = 4kk+2g, k1 = k0+1)  [kk = 0,1].
        v2f bre[16], bim[16];
#pragma unroll
        for (int kk = 0; kk < 2; ++kk) {
            const int k0 = 4 * kk + 2 * g;
            // f2[{k0,k1}, x, r, m] packed as v2f over the member axis
            v2f b_re[3], b_im[3];
            const int o = k0 * FSTRIDE + x * 9 + m;
#pragma unroll
            for (int r = 0; r < 3; ++r) {
                b_re[r] = (v2f){ f2re[o + r * 3], f2re[o + FSTRIDE + r * 3] };
                b_im[r] = (v2f){ f2im[o + r * 3], f2im[o + FSTRIDE + r * 3] };
            }
#pragma unroll
            for (int j = 0; j < 8; ++j) {
                v2f acc_re = {0.0f, 0.0f};
                v2f acc_im = {0.0f, 0.0f};
#pragma unroll
                for (int r = 0; r < 3; ++r) {
                    const v2f ar = { a_re[j][r], a_re[j][r] };   // op_sel broadcast
                    const v2f ai = { a_im[j][r], a_im[j][r] };
                    acc_re += ar * b_re[r];
                    acc_re += (v2f){ -a_im[j][r], -a_im[j][r] } * b_im[r];
                    acc_im += ar * b_im[r];
                    acc_im += ai * b_re[r];
                }
                bre[2 * j + kk] = acc_re;   // == B frag for K-step t, rows P_re
                bim[2 * j + kk] = acc_im;   // == B frag for K-step t+16, rows P_im
            }
        }

        // ---- Stage 2: C[16,16] += A'[16,128] @ B'[128,16], 32 WMMAs --------
        v8f c8 = {};
#pragma unroll
        for (int t = 0; t < 16; ++t) {                 // K rows 0..63 : P_re
            c8 = __builtin_amdgcn_wmma_f32_16x16x4_f32(false, af[t], false, bre[t],
                                                       (short)0, c8, false, false);
        }
#pragma unroll
        for (int t = 0; t < 16; ++t) {                 // K rows 64..127 : P_im
            c8 = __builtin_amdgcn_wmma_f32_16x16x4_f32(false, af[t + 16], false, bim[t],
                                                       (short)0, c8, false, false);
        }

        // ---- Store: C VGPR v -> row M = v + 8*g, column N = r16 == column c.
        // Row M -> out[p = M>>3][i = M&7], flat offset (p*8+i)*NCOLS + c.
#pragma unroll
        for (int v = 0; v < 8; ++v) {
            out[(size_t)(g * 8 + v) * (size_t)NCOLS + (size_t)c] = c8[v];
        }
    }
}

extern "C" void kernel_launch(void* const* d_in, const int* in_sizes, int n_in,
                              void* d_out, int out_size, void* d_ws, size_t ws_size,
                              hipStream_t stream) {
    const float* f1re = (const float*)d_in[0];
    const float* f1im = (const float*)d_in[1];
    const float* f2re = (const float*)d_in[2];
    const float* f2im = (const float*)d_in[3];
    const float* wre  = (const float*)d_in[4];
    const float* wim  = (const float*)d_in[5];
    float* out = (float*)d_out;

    // 73,728 tiles; 4 tiles per wave, 8 waves per 256-thread block.
    dim3 grid(NTILES / (WPB * TPW));   // 2304 blocks
    dim3 block(256);
    lge_bilinear_wmma<<<grid, block, 0, stream>>>(f1re, f1im, f2re, f2im, wre, wim, out);
}